// MessagePassingModule_21165598835458
// MI455X (gfx1250) — compile-verified
//
#include <hip/hip_runtime.h>
#include <stdint.h>

// Problem constants (from reference setup_inputs)
#define B_ 4
#define H_ 256
#define W_ 512
#define L_ 21
#define F_ (B_*H_*W_*L_)          // 11,010,048 elements per field
#define N4_ (F_/4)                // float4 count per field
#define MAX_ITER 5
#define GAMMA 2.0f

#define PF 6                      // async prefetch depth (ring slots)
#define WPB 8                     // waves per block
#define LINES01 (B_*H_)           // 1024 horizontal lines (per dir)
#define LINES23 (B_*W_)           // 2048 vertical lines (per dir)
#define NLINES (2*LINES01 + 2*LINES23)   // 6144 (dir,line) waves

__device__ __forceinline__ float readlane_f(float v, int l) {
  return __uint_as_float((unsigned)__builtin_amdgcn_readlane((int)__float_as_uint(v), l));
}

// min over lanes 0..20 (lanes 21..31 padded with +inf), xor-butterfly over wave32
__device__ __forceinline__ float wave_min21(float v, int lane) {
  float x = (lane < L_) ? v : __builtin_inff();
  x = fminf(x, __shfl_xor(x, 16, 32));
  x = fminf(x, __shfl_xor(x, 8, 32));
  x = fminf(x, __shfl_xor(x, 4, 32));
  x = fminf(x, __shfl_xor(x, 2, 32));
  x = fminf(x, __shfl_xor(x, 1, 32));
  return x;
}

// CDNA5 async copy: per-lane 4B global -> LDS, GVS addressing (SGPR base + 32b voffset).
// Tracked by ASYNCcnt.
__device__ __forceinline__ void async_one(unsigned la, unsigned voff, unsigned long long b) {
  asm volatile("global_load_async_to_lds_b32 %0, %1, %2"
               :: "v"(la), "v"(voff), "s"(b) : "memory");
}

// One wave per (direction, scanline); lane = label. FIRST: msgs==0 => u = total(=phi),
// single prefetch stream, no msg_old reads.
template <bool FIRST>
__global__ __launch_bounds__(256) void k_sweep(const float* __restrict__ total,
                                               float* __restrict__ msgs,
                                               const float* __restrict__ ctx) {
  __shared__ float2 sbuf[WPB][PF][32];   // 12 KB: per-wave ring of (total, msg_old) rows

  const int lane = threadIdx.x & 31;
  const int wv   = threadIdx.x >> 5;
  const int g    = blockIdx.x * WPB + wv;
  if (g >= NLINES) return;

  int d, S, stride, p0;
  if (g < LINES01) {                    // dir 0: L->R along W
    d = 0; S = W_; stride = L_;
    p0 = g * (W_*L_);
  } else if (g < 2*LINES01) {           // dir 1: R->L along W
    d = 1; S = W_; stride = -L_;
    int line = g - LINES01;
    p0 = line * (W_*L_) + (W_-1)*L_;
  } else if (g < 2*LINES01 + LINES23) { // dir 2: T->B along H
    d = 2; S = H_; stride = W_*L_;
    int line = g - 2*LINES01;
    int b = line >> 9, w = line & 511;
    p0 = b * (H_*W_*L_) + w * L_;
  } else {                              // dir 3: B->T along H
    d = 3; S = H_; stride = -(W_*L_);
    int line = g - 2*LINES01 - LINES23;
    int b = line >> 9, w = line & 511;
    p0 = b * (H_*W_*L_) + (H_-1)*(W_*L_) + w * L_;
  }

  // Load ctx column for this lane; detect uniform-Potts structure (zero diag, c>=0 off-diag)
  const int lc = (lane < L_) ? lane : (L_-1);
  const float c = ctx[d*L_*L_ + 1];     // ctx[d][0][1]
  float ctxT[L_];
  bool rowok = true;
  #pragma unroll
  for (int l = 0; l < L_; ++l) {
    ctxT[l] = ctx[d*L_*L_ + l*L_ + lc];
    rowok &= (ctxT[l] == ((l == lc) ? 0.0f : c));
  }
  unsigned long long bal = __ballot((lane >= L_) || rowok);
  const bool potts = (((unsigned)bal) == 0xFFFFFFFFu) && (c >= 0.0f);

  float* __restrict__ msgsd = msgs + (size_t)d * F_;
  const unsigned long long tb = (unsigned long long)(uintptr_t)total;
  const unsigned long long mb = (unsigned long long)(uintptr_t)msgsd;
  const unsigned sb = (unsigned)(stride * 4);            // signed byte stride (mod 2^32)
  unsigned voffL = (unsigned)((p0 + lane) * 4);          // byte offset of next prefetch row
  const unsigned ldsbase = (unsigned)(uintptr_t)(&sbuf[wv][0][0]);  // low 32b = LDS offset
  const unsigned laneoff = (unsigned)(lane * 8);

  // Prologue: fill the ring PF rows deep (ASYNCcnt = PF or 2*PF)
  #pragma unroll
  for (int j = 0; j < PF; ++j) {
    if (lane < L_) {
      unsigned la = ldsbase + (unsigned)(j*256) + laneoff;
      async_one(la, voffL, tb);
      if (!FIRST) async_one(la + 4u, voffL, mb);
    }
    voffL += sb;
  }

  int pcur = p0;
  float m = 0.0f;
  float u_prev;

  // step 0: m_0 = 0; consume row 0, reissue slot 0 for row PF
  {
    if (FIRST) asm volatile("s_wait_asynccnt 5"  ::: "memory");  // PF-1
    else       asm volatile("s_wait_asynccnt 10" ::: "memory");  // 2*(PF-1)
    float2 v = sbuf[wv][0][lane];
    u_prev = FIRST ? v.x : (v.x - v.y);                  // u = total - msg_old
    if (lane < L_) {
      msgsd[pcur + lane] = 0.0f;
      asm volatile("s_wait_dscnt 0" ::: "memory");       // LDS read of slot 0 retired
      async_one(ldsbase + laneoff, voffL, tb);
      if (!FIRST) async_one(ldsbase + laneoff + 4u, voffL, mb);
    }
    voffL += sb;
    pcur  += stride;
  }

  int slot = 1;
  for (int i = 1; i < S; ++i) {
    if (FIRST) asm volatile("s_wait_asynccnt 5"  ::: "memory");
    else       asm volatile("s_wait_asynccnt 10" ::: "memory");
    float2 v = sbuf[wv][slot][lane];
    float u = FIRST ? v.x : (v.x - v.y);

    // recurrence: m_i(l') = min_l( u_{i-1}(l) + m_{i-1}(l) + ctx(l,l') ), normalized
    float t = u_prev + m;
    if (potts) {
      float tmin = wave_min21(t, lane);
      m = fminf(t - tmin, c);                            // == min(t, tmin+c) - tmin
    } else {
      float a0 = __builtin_inff(), a1 = a0, a2 = a0;
      #pragma unroll
      for (int l = 0; l < L_; ++l) {
        float cand = readlane_f(t, l) + ctxT[l];
        if ((l % 3) == 0)      a0 = fminf(a0, cand);
        else if ((l % 3) == 1) a1 = fminf(a1, cand);
        else                   a2 = fminf(a2, cand);
      }
      float mn = fminf(a0, fminf(a1, a2));
      m = mn - wave_min21(mn, lane);
    }

    if (lane < L_) {
      msgsd[pcur + lane] = m;                            // in-place (old value consumed)
      if (i + PF < S) {
        asm volatile("s_wait_dscnt 0" ::: "memory");     // slot's LDS read retired before reuse
        unsigned la = ldsbase + (unsigned)(slot*256) + laneoff;
        async_one(la, voffL, tb);
        if (!FIRST) async_one(la + 4u, voffL, mb);
      }
    }
    voffL += sb;
    pcur  += stride;
    u_prev = u;
    slot = (slot + 1 == PF) ? 0 : slot + 1;
  }
}

__global__ __launch_bounds__(256) void k_total(const float4* __restrict__ phi,
                                               const float4* __restrict__ msgs,
                                               float4* __restrict__ tot) {
  int i = blockIdx.x * 256 + threadIdx.x;
  if (i >= N4_) return;
  float4 p  = phi[i];
  float4 m0 = msgs[i], m1 = msgs[i + N4_], m2 = msgs[i + 2*N4_], m3 = msgs[i + 3*N4_];
  float4 r;
  r.x = p.x + m0.x + m1.x + m2.x + m3.x;
  r.y = p.y + m0.y + m1.y + m2.y + m3.y;
  r.z = p.z + m0.z + m1.z + m2.z + m3.z;
  r.w = p.w + m0.w + m1.w + m2.w + m3.w;
  tot[i] = r;
}

__global__ __launch_bounds__(256) void k_final(const float4* __restrict__ phi,
                                               const float4* __restrict__ msgs,
                                               float4* __restrict__ out) {
  int i = blockIdx.x * 256 + threadIdx.x;
  if (i >= N4_) return;
  float4 p  = phi[i];
  float4 m0 = msgs[i], m1 = msgs[i + N4_], m2 = msgs[i + 2*N4_], m3 = msgs[i + 3*N4_];
  float4 r;
  r.x = GAMMA * p.x + m0.x + m1.x + m2.x + m3.x;
  r.y = GAMMA * p.y + m0.y + m1.y + m2.y + m3.y;
  r.z = GAMMA * p.z + m0.z + m1.z + m2.z + m3.z;
  r.w = GAMMA * p.w + m0.w + m1.w + m2.w + m3.w;
  out[i] = r;
}

extern "C" void kernel_launch(void* const* d_in, const int* in_sizes, int n_in,
                              void* d_out, int out_size, void* d_ws, size_t ws_size,
                              hipStream_t stream) {
  (void)in_sizes; (void)n_in; (void)out_size; (void)ws_size;
  const float* phi = (const float*)d_in[0];
  const float* ctx = (const float*)d_in[1];   // [4][21][21]
  float* msgs  = (float*)d_ws;                // 4 * F_ floats (176 MB), updated in place
  float* total = msgs + (size_t)4 * F_;       // F_ floats (44 MB)
  float* out   = (float*)d_out;

  // Iteration 0: msgs == 0  =>  total == phi, u == phi. Streams only phi, writes all of
  // msgs (every line position), so no zero-init pass and no aggregate pass are needed.
  k_sweep<true><<<NLINES/WPB, 256, 0, stream>>>(phi, msgs, ctx);

  for (int it = 1; it < MAX_ITER; ++it) {
    k_total<<<N4_/256, 256, 0, stream>>>((const float4*)phi, (const float4*)msgs,
                                         (float4*)total);
    k_sweep<false><<<NLINES/WPB, 256, 0, stream>>>(total, msgs, ctx);
  }
  k_final<<<N4_/256, 256, 0, stream>>>((const float4*)phi, (const float4*)msgs,
                                       (float4*)out);
}